// Mask_33243046871374
// MI455X (gfx1250) — compile-verified
//
#include <hip/hip_runtime.h>
#include <stdint.h>

// Problem constants (reference: shape [2,4,2048,2048], k=20, eps=1e-7)
#define NN        2048
#define KSEL      20
#define LBATCH    8                 // L*B = 2*4
#define NROWS     (LBATCH * NN)     // 16384 row-topk problems
#define EPSF      1e-7f
#define TPB_TOPK  256
#define EPT       8                 // NN / TPB_TOPK elements per thread

// Native clang vector (layout-compatible with float4, accepted by
// __builtin_nontemporal_store / vectorized b128 memory ops).
typedef float f32x4 __attribute__((ext_vector_type(4)));

// ---------------- CDNA5 async global->LDS helpers (gfx1250) ----------------
// GVS mode: the global address is the 64-bit base in the saddr SGPR pair plus the 32-bit byte offset in vaddr;
// 16 bytes are copied to the LDS address in vdst. Tracked by ASYNCcnt.
__device__ __forceinline__ void async_load_b128(unsigned lds_off, unsigned voff,
                                                const void* sbase) {
  asm volatile("global_load_async_to_lds_b128 %0, %1, %2"
               :
               : "v"(lds_off), "v"(voff), "s"(sbase)
               : "memory");
}
// Non-temporal flavor: single-use stream (keeps adj resident in the 192MB L2).
__device__ __forceinline__ void async_load_b128_nt(unsigned lds_off,
                                                   unsigned voff,
                                                   const void* sbase) {
  asm volatile("global_load_async_to_lds_b128 %0, %1, %2 th:TH_LOAD_NT"
               :
               : "v"(lds_off), "v"(voff), "s"(sbase)
               : "memory");
}
__device__ __forceinline__ void wait_asynccnt0() {
  asm volatile("s_wait_asynccnt 0" ::: "memory");
}

// wave32 64-bit xor-shuffle built from two 32-bit shuffles
__device__ __forceinline__ unsigned long long shfl_xor_u64(unsigned long long v,
                                                           int lanemask) {
  unsigned lo = (unsigned)(v & 0xffffffffull);
  unsigned hi = (unsigned)(v >> 32);
  lo = __shfl_xor(lo, lanemask, 32);
  hi = __shfl_xor(hi, lanemask, 32);
  return (((unsigned long long)hi) << 32) | lo;
}

// Monotone total-order key for floats (neg -> ~bits, nonneg -> bits|0x80000000)
__device__ __forceinline__ unsigned fkey(float f) {
  unsigned u = __float_as_uint(f);
  return u ^ (((unsigned)((int)u >> 31)) | 0x80000000u);
}

// ---------------------------------------------------------------------------
__global__ void k_zero_flags(unsigned* __restrict__ flags) {
  int i = blockIdx.x * blockDim.x + threadIdx.x;
  if (i < NROWS) flags[i] = 0u;
}

// One block per row: exact top-20 indices of |adj|+0.01*noise over 2048 cols.
// Sets bit j of flags[lb*NN + idx_j] (the dim-1 scatter of the reference).
__global__ __launch_bounds__(TPB_TOPK)
void k_topk(const float* __restrict__ adj, const float* __restrict__ noise,
            unsigned* __restrict__ flags) {
  __shared__ float smA[NN];
  __shared__ float smB[NN];
  __shared__ unsigned long long red[2][TPB_TOPK / 32];  // double-buffered

  const unsigned t   = threadIdx.x;
  const unsigned row = blockIdx.x;       // = lb*NN + i  (flattened row id)
  const unsigned lb  = row >> 11;        // row / NN
  const unsigned fb  = lb << 11;         // flags base = lb*NN

  // Stage both 8KB rows into LDS via CDNA5 async global->LDS copies.
  // adj: regular-temporal (re-read by k_apply; fits the 192MB L2 alone).
  // noise: non-temporal (single use; don't evict adj).
  const unsigned ldsA = (unsigned)(uintptr_t)(&smA[0]);
  const unsigned ldsB = (unsigned)(uintptr_t)(&smB[0]);
  const unsigned gb   = row * (NN * 4u) + t * 16u;  // byte offset, fits 32 bits
  async_load_b128   (ldsA + t * 16u,         gb,         adj);
  async_load_b128   (ldsA + 4096u + t * 16u, gb + 4096u, adj);
  async_load_b128_nt(ldsB + t * 16u,         gb,         noise);
  async_load_b128_nt(ldsB + 4096u + t * 16u, gb + 4096u, noise);
  wait_asynccnt0();
  __syncthreads();

  const int c0 = (int)t * EPT;
  float s[EPT];
#pragma unroll
  for (int u = 0; u < EPT; ++u)
    s[u] = fabsf(smA[c0 + u]) + 0.01f * smB[c0 + u];

  for (int k = 0; k < KSEL; ++k) {
    // local argmax over private 8 (scores >= 0; removed entries are -1.0f)
    float bv = s[0];
    int bi = c0;
#pragma unroll
    for (int u = 1; u < EPT; ++u)
      if (s[u] > bv) { bv = s[u]; bi = c0 + u; }

    // value-major, lowest-index-tiebreak key (matches lax.top_k stability)
    unsigned long long key =
        (((unsigned long long)fkey(bv)) << 32) | (unsigned)(~bi);

    // wave32 reduce
#pragma unroll
    for (int m = 16; m >= 1; m >>= 1) {
      unsigned long long o = shfl_xor_u64(key, m);
      if (o > key) key = o;
    }
    const int buf = k & 1;
    if ((t & 31u) == 0u) red[buf][t >> 5] = key;
    __syncthreads();  // single barrier per round: next round writes buf^1

    unsigned long long w = red[buf][0];
#pragma unroll
    for (int q = 1; q < TPB_TOPK / 32; ++q) {
      unsigned long long o = red[buf][q];
      if (o > w) w = o;
    }
    const int widx = (int)(~((unsigned)w));  // winning column in [0,2048)

    if (t == 0) atomicOr(&flags[fb + (unsigned)widx], 1u << k);
    if ((unsigned)(widx >> 3) == t) s[widx & 7] = -1.0f;  // remove winner
  }
}

// out = adj * (eps + mask); only cols < 20 can have mask==1 (flags bit c).
__global__ void k_apply(const f32x4* __restrict__ adj4,
                        const unsigned* __restrict__ flags,
                        f32x4* __restrict__ out4) {
  const float mEps = EPSF;
  const float mHit = 1.0f + EPSF;
  const unsigned i   = blockIdx.x * blockDim.x + threadIdx.x;  // < NROWS*512
  const unsigned row = i >> 9;    // / (NN/4)
  const unsigned c4  = i & 511u;
  f32x4 a = adj4[i];
  f32x4 r;
  if (c4 < (KSEL + 3) / 4) {      // c in [0,20): consult flags word
    const unsigned f = flags[row];
    const unsigned c = c4 * 4u;
    r.x = a.x * (((f >> (c + 0)) & 1u) ? mHit : mEps);
    r.y = a.y * (((f >> (c + 1)) & 1u) ? mHit : mEps);
    r.z = a.z * (((f >> (c + 2)) & 1u) ? mHit : mEps);
    r.w = a.w * (((f >> (c + 3)) & 1u) ? mHit : mEps);
  } else {
    r.x = a.x * mEps; r.y = a.y * mEps; r.z = a.z * mEps; r.w = a.w * mEps;
  }
  __builtin_nontemporal_store(r, &out4[i]);  // output never re-read
}

// ---------------------------------------------------------------------------
extern "C" void kernel_launch(void* const* d_in, const int* in_sizes, int n_in,
                              void* d_out, int out_size, void* d_ws,
                              size_t ws_size, hipStream_t stream) {
  (void)in_sizes; (void)n_in; (void)out_size; (void)ws_size;
  const float* adj   = (const float*)d_in[0];
  const float* noise = (const float*)d_in[1];
  float* out         = (float*)d_out;
  unsigned* flags    = (unsigned*)d_ws;  // NROWS u32 = 64 KB scratch

  k_zero_flags<<<NROWS / 256, 256, 0, stream>>>(flags);
  k_topk<<<NROWS, TPB_TOPK, 0, stream>>>(adj, noise, flags);

  const unsigned total4 = NROWS * (NN / 4);  // 8,388,608 float4s
  k_apply<<<total4 / 256, 256, 0, stream>>>((const f32x4*)adj, flags,
                                            (f32x4*)out);
}